// AnalyticalBoundedLineAttractor_26027501814666
// MI455X (gfx1250) — compile-verified
//
#include <hip/hip_runtime.h>

typedef __attribute__((ext_vector_type(2))) float v2f;
typedef __attribute__((ext_vector_type(8))) float v8f;

#define DIM     64
#define TSTEPS  100
#define DT      0.05f
#define KTERMS  8          // Taylor terms of phi-series; ||A*dt||~0.15 -> ~1e-12
#define MTILE   16
#define LSTR    68         // padded LDS row stride (floats), bank-conflict free

__global__ __launch_bounds__(128)
void AnalyticalBoundedLineAttractor_kernel(const float* __restrict__ x0,
                                           const float* __restrict__ W,
                                           const float* __restrict__ b,
                                           float* __restrict__ out)
{
    __shared__ float xbuf[MTILE * LSTR];
    __shared__ float tbuf[2][MTILE * LSTR];

    const int tid  = threadIdx.x;
    const int wave = tid >> 5;            // 0..3 : output column block
    const int lane = tid & 31;
    const int ln   = lane & 15;
    const int hi   = lane >> 4;
    const int nb   = wave * 16;           // column base
    const int mb   = blockIdx.x * MTILE;  // batch row base

    // ---- Preload B operand: W^T as (K x N) fragments, 16 k-tiles, kept in VGPRs ----
    // B[k][n] = W[nb+n][k]; lane<16 holds k=k0,k0+1, lane>=16 holds k=k0+2,k0+3.
    v2f wfrag[16];
    {
        const int r = nb + ln;            // W row == output column
        #pragma unroll
        for (int kt = 0; kt < 16; ++kt) {
            const int kk = kt * 4 + hi * 2;
            wfrag[kt].x = W[r * DIM + kk];
            wfrag[kt].y = W[r * DIM + kk + 1];
        }
    }
    const float bn = b[nb + ln];          // per-lane bias (column is fixed per lane)

    // ---- Load initial X tile (16 x 64) into LDS ----
    for (int i = tid; i < MTILE * DIM; i += 128) {
        const int m = i >> 6, d = i & 63;
        xbuf[m * LSTR + d] = x0[(mb + m) * DIM + d];
    }
    __syncthreads();

    // Lane-private x at this lane's C/D-fragment positions: (m = v+8*hi, n = nb+ln)
    v8f xv;
    #pragma unroll
    for (int v = 0; v < 8; ++v)
        xv[v] = xbuf[(v + 8 * hi) * LSTR + nb + ln];

    for (int t = 0; t < TSTEPS; ++t) {
        // ---- Trajectory stores state BEFORE the update ----
        #pragma unroll
        for (int v = 0; v < 8; ++v) {
            const int m = v + 8 * hi;
            out[(size_t)(mb + m) * (TSTEPS * DIM) + (size_t)t * DIM + nb + ln] = xv[v];
        }

        // ---- Wx = X @ W^T  (16 fp32 WMMAs over K=64) ----
        v8f acc = {};
        #pragma unroll
        for (int kt = 0; kt < 16; ++kt) {
            const int kk = kt * 4 + hi * 2;
            v2f a;
            a.x = xbuf[ln * LSTR + kk];
            a.y = xbuf[ln * LSTR + kk + 1];
            acc = __builtin_amdgcn_wmma_f32_16x16x4_f32(false, a, false, wfrag[kt],
                                                        (short)0, acc, false, false);
        }

        // ---- Regime mask + first series term: term1 = dt*(relu(z) - x) ----
        v8f mask, termv, s;
        #pragma unroll
        for (int v = 0; v < 8; ++v) {
            const float z  = acc[v] + bn;          // pre-activation
            const float mk = (z > 0.0f) ? 1.0f : 0.0f;
            mask[v]  = mk;
            const float t1 = DT * (mk * z - xv[v]); // m.z - x == A x + b_eff
            termv[v] = t1;
            s[v]     = xv[v] + t1;
        }
        #pragma unroll
        for (int v = 0; v < 8; ++v)
            tbuf[0][(v + 8 * hi) * LSTR + nb + ln] = termv[v];
        __syncthreads();

        // ---- Higher terms: term_{k} = (dt/k)*(m.(W term) - term) ----
        int p = 0;
        #pragma unroll
        for (int k = 2; k <= KTERMS; ++k) {
            v8f wt = {};
            #pragma unroll
            for (int kt = 0; kt < 16; ++kt) {
                const int kk = kt * 4 + hi * 2;
                v2f a;
                a.x = tbuf[p][ln * LSTR + kk];
                a.y = tbuf[p][ln * LSTR + kk + 1];
                wt = __builtin_amdgcn_wmma_f32_16x16x4_f32(false, a, false, wfrag[kt],
                                                           (short)0, wt, false, false);
            }
            const float c = DT / (float)k;
            #pragma unroll
            for (int v = 0; v < 8; ++v) {
                const float nt = c * (mask[v] * wt[v] - termv[v]);
                termv[v] = nt;
                s[v]    += nt;
            }
            if (k < KTERMS) {
                #pragma unroll
                for (int v = 0; v < 8; ++v)
                    tbuf[p ^ 1][(v + 8 * hi) * LSTR + nb + ln] = termv[v];
            }
            __syncthreads();
            p ^= 1;
        }

        // ---- Commit x_next ----
        xv = s;
        #pragma unroll
        for (int v = 0; v < 8; ++v)
            xbuf[(v + 8 * hi) * LSTR + nb + ln] = xv[v];
        __syncthreads();
    }
}

extern "C" void kernel_launch(void* const* d_in, const int* in_sizes, int n_in,
                              void* d_out, int out_size, void* d_ws, size_t ws_size,
                              hipStream_t stream) {
    const float* x0 = (const float*)d_in[0];   // (256, 64)
    const float* W  = (const float*)d_in[1];   // (64, 64)
    const float* b  = (const float*)d_in[2];   // (64,)
    float* out = (float*)d_out;                // (256, 100, 64)

    const int batch = in_sizes[0] / DIM;       // 256
    dim3 grid(batch / MTILE);                  // 16 workgroups
    dim3 block(128);                           // 4 waves (wave32)
    AnalyticalBoundedLineAttractor_kernel<<<grid, block, 0, stream>>>(x0, W, b, out);
}